// LabelLoss_89936615178266
// MI455X (gfx1250) — compile-verified
//
#include <hip/hip_runtime.h>
#include <stdint.h>

// Problem dims (fixed by the reference)
#define B_  16
#define K_  11
#define H_  128
#define W_  128
#define HW_ (H_ * W_)
#define C7  7
#define NT  256   // threads per block (8 wave32s)

typedef unsigned int v4u __attribute__((ext_vector_type(4)));
typedef int          v4i __attribute__((ext_vector_type(4)));
typedef int          v8i __attribute__((ext_vector_type(8)));

#if __has_builtin(__builtin_amdgcn_tensor_load_to_lds)
#define HAS_TDM 1
#else
#define HAS_TDM 0
#endif

#if HAS_TDM
// One TDM DMA: load a 128x128 f32 tile (row-major, stride 128) from global into LDS.
// Descriptor layout per CDNA5 ISA ch.8 (D# groups 0/1; groups 2/3 unused for 2D).
__device__ __forceinline__ void tdm_load_tile_128x128_f32(const float* gsrc, void* lds_dst) {
  unsigned lds_off = (unsigned)(uintptr_t)lds_dst;          // low 32 bits of generic ptr == LDS offset
  unsigned long long ga = (unsigned long long)(uintptr_t)gsrc;

  v4u g0;
  g0.x = 1u;                                                // count=1 (valid), is_restore=0, gather off
  g0.y = lds_off;                                           // lds_addr (bytes)
  g0.z = (unsigned)(ga & 0xFFFFFFFFull);                    // global_addr[31:0]
  g0.w = (unsigned)(((ga >> 32) & 0x01FFFFFFull)            // global_addr[56:32]
                    | 0x80000000u);                         // type = 2 ("image")

  v8i g1;
  g1[0] = 0x00020000;          // workgroup_mask=0, data_size=2 (4 bytes), no barrier/iterate/pad
  g1[1] = (int)(128u << 16);   // tensor_dim0[15:0] = 128 (bits 63:48)
  g1[2] = (int)(128u << 16);   // tensor_dim0 hi = 0, tensor_dim1[15:0] = 128 (bits 95:80)
  g1[3] = (int)(128u << 16);   // tensor_dim1 hi = 0, tile_dim0 = 128 (bits 127:112)
  g1[4] = 128;                 // tile_dim1 = 128, tile_dim2 = 0
  g1[5] = 128;                 // tensor_dim0_stride = 128 elements (low 32 bits)
  g1[6] = 0;                   // stride hi, tensor_dim1_stride lo = 0 (unused for 2D)
  g1[7] = 0;

  v4i zero4 = {0, 0, 0, 0};
#if defined(__clang_major__) && (__clang_major__ >= 23)
  v8i zero8 = {0, 0, 0, 0, 0, 0, 0, 0};
  __builtin_amdgcn_tensor_load_to_lds(g0, g1, zero4, zero4, zero8, 0);
#else
  __builtin_amdgcn_tensor_load_to_lds(g0, g1, zero4, zero4, 0);
#endif
  __builtin_amdgcn_s_wait_tensorcnt(0);                     // drain TENSORcnt before barrier
}
#endif

// One workgroup per (b,k): pooled 3x3 argmax over 128x128 map + 7-channel gather loss.
__global__ __launch_bounds__(NT) void pooled_argmax_loss_kernel(
    const float* __restrict__ pred,     // [B, K*7, H, W]
    const float* __restrict__ gt,       // [B, K, 7]
    const float* __restrict__ heatmap,  // [B, K, H, W]
    float* __restrict__ loss_bk) {      // [B*K] workspace
  __shared__ float sh[HW_];             // 64 KB tile (reused for the reduction tail)

  const int bk  = blockIdx.x;
  const int b   = bk / K_;
  const int k   = bk % K_;
  const int tid = threadIdx.x;
  const float* src = heatmap + (size_t)bk * HW_;

#if HAS_TDM
  if (tid < 32) {                       // wave 0 issues the DMA + waits TENSORcnt
    tdm_load_tile_128x128_f32(src, sh);
  }
#else
  for (int i = tid; i < HW_ / 4; i += NT) {
    ((float4*)sh)[i] = ((const float4*)src)[i];
  }
#endif
  __syncthreads();

  // Zero-padded 3x3 window sums (uniform /9 dropped: argmax-invariant) + local argmax.
  // Iterating p in increasing order with strict '>' keeps the FIRST max per thread.
  float bestV = -INFINITY;
  int   bestI = 0;
  for (int p = tid; p < HW_; p += NT) {
    const int r = p >> 7;
    const int c = p & (W_ - 1);
    float s = 0.0f;
#pragma unroll
    for (int dr = -1; dr <= 1; ++dr) {
      const int rr = r + dr;
      if ((unsigned)rr < (unsigned)H_) {
        const float* row = sh + rr * W_;
#pragma unroll
        for (int dc = -1; dc <= 1; ++dc) {
          const int cc = c + dc;
          if ((unsigned)cc < (unsigned)W_) s += row[cc];
        }
      }
    }
    if (s > bestV) { bestV = s; bestI = p; }
  }
  __syncthreads();                      // everyone done reading sh; reuse it

  float* sVal = sh;                     // 256 floats
  int*   sIdx = (int*)(sh + NT);        // 256 ints
  sVal[tid] = bestV;
  sIdx[tid] = bestI;
  __syncthreads();

  // Tree reduction; tie-break to the smaller flat index (jnp.argmax semantics).
  for (int stride = NT / 2; stride > 0; stride >>= 1) {
    if (tid < stride) {
      const float v2 = sVal[tid + stride];
      const int   i2 = sIdx[tid + stride];
      if (v2 > sVal[tid] || (v2 == sVal[tid] && i2 < sIdx[tid])) {
        sVal[tid] = v2;
        sIdx[tid] = i2;
      }
    }
    __syncthreads();
  }

  if (tid == 0) {
    const int idx = sIdx[0];
    const int x = idx >> 7;             // index // 128 (row)
    const int y = idx & (W_ - 1);       // index %  128 (col)
    const float* pbase = pred + (((size_t)b * (K_ * C7) + k * C7) * H_ + x) * W_ + y;
    const float* gbase = gt + ((size_t)b * K_ + k) * C7;
    float acc = 0.0f;
#pragma unroll
    for (int c = 0; c < C7; ++c) {
      const float d = pbase[(size_t)c * HW_] - gbase[c];
      acc = fmaf(d, d, acc);
    }
    loss_bk[bk] = acc;                  // MSE(reduction='sum') per (b,k)
  }
}

// out[b] = mean over K of loss_bk[b,k]  (deterministic, no atomics)
__global__ void mean_over_k_kernel(const float* __restrict__ loss_bk,
                                   float* __restrict__ out) {
  const int b = threadIdx.x;
  if (b < B_) {
    float s = 0.0f;
#pragma unroll
    for (int k = 0; k < K_; ++k) s += loss_bk[b * K_ + k];
    out[b] = s * (1.0f / (float)K_);
  }
}

extern "C" void kernel_launch(void* const* d_in, const int* in_sizes, int n_in,
                              void* d_out, int out_size, void* d_ws, size_t ws_size,
                              hipStream_t stream) {
  const float* pred    = (const float*)d_in[0];   // [16, 77, 128, 128]
  const float* gt      = (const float*)d_in[1];   // [16, 11, 7]
  const float* heatmap = (const float*)d_in[2];   // [16, 11, 128, 128]
  float* out     = (float*)d_out;                 // [16]
  float* loss_bk = (float*)d_ws;                  // 176 floats of scratch

  pooled_argmax_loss_kernel<<<B_ * K_, NT, 0, stream>>>(pred, gt, heatmap, loss_bk);
  mean_over_k_kernel<<<1, 32, 0, stream>>>(loss_bk, out);
}